// MADPF_44890998178253
// MI455X (gfx1250) — compile-verified
//
#include <hip/hip_runtime.h>
#include <math.h>

#define BB 32
#define NN 16384
#define DD 8
#define MM 8

typedef float v2f __attribute__((ext_vector_type(2)));
typedef float v8f __attribute__((ext_vector_type(8)));

// ---------------------------------------------------------------------------
// Kernel 1: per-batch segment sizes.  spm[b,m], cum[b,m] from softmax of
// model_posteriors.  Tiny (B=32, M=8): one wave per b, serial on lane 0.
// ---------------------------------------------------------------------------
__global__ void k_segments(const float* __restrict__ model_posteriors,
                           int* __restrict__ ws_cum, int* __restrict__ ws_spm) {
  int b = blockIdx.x;
  if (threadIdx.x == 0) {
    float mp[MM];
    float mx = -INFINITY;
    for (int m = 0; m < MM; ++m) { mp[m] = model_posteriors[b * MM + m]; mx = fmaxf(mx, mp[m]); }
    float s = 0.f;
    for (int m = 0; m < MM; ++m) s += __expf(mp[m] - mx);
    float lse = mx + __logf(s);
    const int base = NN / (MM * 5);
    int spm[MM];
    int tot = 0;
    for (int m = 0; m < MM; ++m) {
      float p = __expf(mp[m] - lse);
      spm[m] = base + (int)floorf((float)(NN - base * MM) * p);
      tot += spm[m];
    }
    spm[0] += NN - tot;  // rounding fix
    int c = 0;
    for (int m = 0; m < MM; ++m) {
      c += spm[m];
      ws_cum[b * MM + m] = c;
      ws_spm[b * MM + m] = spm[m];
    }
  }
}

// ---------------------------------------------------------------------------
// Kernel 2: per (m,b) masked softmax CDF over N=16384 particles.
//   pass A: masked max      pass B: masked sum-exp -> lse
//   pass C: inclusive prefix sum of probs using chained f32 WMMA:
//           a 16-elem scan is L(16x16, lower-tri ones) * X; we build the
//           K=16 product from 4 chained V_WMMA_F32_16X16X4_F32 ops.
//           Each wave scans 16 columns x 16 elems = 256 elems per WMMA group.
// ---------------------------------------------------------------------------
__global__ void __launch_bounds__(256)
k_cdf(const float* __restrict__ w,     // [B,N] modelwise_w
      const int* __restrict__ midx,    // [B,N] model_idx
      float* __restrict__ cdf)         // [M,B,N]
{
  const int mb = blockIdx.x;  // m*B + b
  const int m = mb / BB;
  const int b = mb % BB;
  const float* wrow = w + (size_t)b * NN;
  const int* irow = midx + (size_t)b * NN;
  float* crow = cdf + (size_t)mb * NN;

  __shared__ float s_red[256];
  __shared__ float s_wsum[8];
  __shared__ float s_carry;
  __shared__ float s_tile[2048];

  const int t = threadIdx.x;

  // ---- pass A: masked max ----
  float mx = -INFINITY;
  for (int n = t; n < NN; n += 256) {
    float wv = wrow[n];
    if (irow[n] == m) mx = fmaxf(mx, wv);
  }
  s_red[t] = mx;
  __syncthreads();
  for (int o = 128; o > 0; o >>= 1) {
    if (t < o) s_red[t] = fmaxf(s_red[t], s_red[t + o]);
    __syncthreads();
  }
  mx = s_red[0];
  __syncthreads();

  // ---- pass B: masked sum-exp ----
  float sm = 0.f;
  for (int n = t; n < NN; n += 256) {
    if (irow[n] == m) sm += __expf(wrow[n] - mx);
  }
  s_red[t] = sm;
  __syncthreads();
  for (int o = 128; o > 0; o >>= 1) {
    if (t < o) s_red[t] += s_red[t + o];
    __syncthreads();
  }
  float lse = mx + __logf(s_red[0]);
  // empty-mask guard: makes the branch-free (exp * mask) form NaN-safe; the
  // mask is all-zero in that case so the CDF is all-zero either way.
  if (!(lse > -INFINITY)) lse = 0.f;

  // ---- pass C: WMMA prefix sum ----
  const int lane = t & 31;
  const int wv = t >> 5;                    // wave id 0..7
  const int col = lane & 15;                // column owned by this lane
  const int koff = (lane < 16) ? 0 : 2;     // K half (A/B layout)
  const int rowoff = (lane < 16) ? 0 : 8;   // D row half

  // hoisted lower-triangular-ones A slices (invariant across tiles)
  v2f amat[4];
#pragma unroll
  for (int j = 0; j < 4; ++j) {
    int k0 = 4 * j + koff;
    amat[j].x = (k0 <= col) ? 1.0f : 0.0f;
    amat[j].y = (k0 + 1 <= col) ? 1.0f : 0.0f;
  }

  if (t == 0) s_carry = 0.f;
  __syncthreads();

  for (int tile = 0; tile < NN; tile += 2048) {
    const int cbase = tile + wv * 256;  // this wave's 256-element chunk
    v8f acc = {0.f, 0.f, 0.f, 0.f, 0.f, 0.f, 0.f, 0.f};
#pragma unroll
    for (int j = 0; j < 4; ++j) {
      // B^(j)[k][n] = prob(element n*16 + 4j + k) of this chunk
      int e = cbase + col * 16 + 4 * j + koff;
      float2 w2 = *(const float2*)(wrow + e);
      int2 i2 = *(const int2*)(irow + e);
      v2f bmat;  // branch-free masked softmax probability
      bmat.x = __expf(w2.x - lse) * ((i2.x == m) ? 1.0f : 0.0f);
      bmat.y = __expf(w2.y - lse) * ((i2.y == m) ? 1.0f : 0.0f);
      acc = __builtin_amdgcn_wmma_f32_16x16x4_f32(false, amat[j], false, bmat,
                                                  (short)0, acc, false, false);
    }
    // acc[r] = inclusive scan value of element  cbase + col*16 + rowoff + r

    // column totals: D[15][col] lives in vgpr7, lane 16+col
    float ctot = __shfl(acc[7], 16 + col, 32);
    // exclusive scan across 16 columns (segmented width-16; both halves equal)
    float inc = ctot;
#pragma unroll
    for (int o = 1; o < 16; o <<= 1) {
      float up = __shfl_up(inc, o, 16);
      if (col >= o) inc += up;
    }
    float cexcl = inc - ctot;
    float wtot = __shfl(inc, 15, 16);  // this wave's chunk total
    if (lane == 0) s_wsum[wv] = wtot;
    __syncthreads();

    float pre = s_carry;
    float allsum = 0.f;
#pragma unroll
    for (int ww = 0; ww < 8; ++ww) {
      float v = s_wsum[ww];
      if (ww < wv) pre += v;
      allsum += v;
    }
    float off = pre + cexcl;
#pragma unroll
    for (int r = 0; r < 8; ++r) {
      s_tile[wv * 256 + col * 16 + rowoff + r] = acc[r] + off;
    }
    __syncthreads();
    if (t == 0) s_carry += allsum;
    // coalesced store of the 2048-element tile
#pragma unroll
    for (int k2 = 0; k2 < 8; ++k2) {
      int idx = k2 * 256 + t;
      crow[tile + idx] = s_tile[idx];
    }
    __syncthreads();
  }
}

// ---------------------------------------------------------------------------
// Kernel 3: per particle: segment -> model, inverse-CDF search, gather x_t,
// and log_w = cond_likelihoods[b, m_new] + obs_loglik[b, n].
// b is uniform per block (N % 256 == 0) -> scalar loads for cum/cond_lik.
// ---------------------------------------------------------------------------
__global__ void __launch_bounds__(256)
k_resample(const float* __restrict__ x_t, const float* __restrict__ u,
           const float* __restrict__ cond_lik, const float* __restrict__ obs,
           const int* __restrict__ ws_cum, const float* __restrict__ cdf,
           int* __restrict__ ws_newidx, float* __restrict__ out_x,
           float* __restrict__ out_logw) {
  const int b = (blockIdx.x << 8) / NN;              // uniform per block
  const int n = (blockIdx.x << 8) - b * NN + threadIdx.x;
  const int gid = b * NN + n;

  int mnew = 0;
#pragma unroll
  for (int mm2 = 0; mm2 < MM; ++mm2) mnew += (n >= ws_cum[b * MM + mm2]) ? 1 : 0;
  if (mnew > MM - 1) mnew = MM - 1;
  ws_newidx[gid] = mnew;

  float q = u[((size_t)mnew * BB + b) * NN + n];
  const float* crow = cdf + ((size_t)mnew * BB + b) * NN;
  // branchless lower_bound over 16384 sorted f32 (CDF rows live in L2)
  int base = 0;
#pragma unroll
  for (int half = NN / 2; half >= 1; half >>= 1) {
    base += (crow[base + half - 1] < q) ? half : 0;
  }
  int idxr = base + ((crow[base] < q) ? 1 : 0);
  int sel = (idxr < NN - 1) ? idxr : NN - 1;

  const float4* src = (const float4*)(x_t + ((size_t)b * NN + sel) * DD);
  float4* dst = (float4*)(out_x + (size_t)gid * DD);
  dst[0] = src[0];
  dst[1] = src[1];

  out_logw[gid] = cond_lik[b * MM + mnew] + obs[gid];
}

// ---------------------------------------------------------------------------
// Kernel 4: per (b,m) segment LSE of obs_loglik; cond_new = cond_l + lseg -
// log(count).  (log_w is constant-offset obs within a segment, so this is
// exactly the reference's masked logsumexp of log_w.)
// ---------------------------------------------------------------------------
__global__ void __launch_bounds__(256)
k_seglse(const float* __restrict__ obs, const float* __restrict__ cond_lik,
         const int* __restrict__ ws_cum, const int* __restrict__ ws_spm,
         float* __restrict__ ws_lse2, float* __restrict__ out_cond) {
  int mb = blockIdx.x;  // b*M + m
  int b = mb / MM;
  int m = mb % MM;
  int end = ws_cum[b * MM + m];
  int start = end - ws_spm[b * MM + m];
  const float* orow = obs + (size_t)b * NN;
  __shared__ float s_red[256];
  int t = threadIdx.x;

  float mx = -INFINITY;
  for (int n = start + t; n < end; n += 256) mx = fmaxf(mx, orow[n]);
  s_red[t] = mx;
  __syncthreads();
  for (int o = 128; o > 0; o >>= 1) {
    if (t < o) s_red[t] = fmaxf(s_red[t], s_red[t + o]);
    __syncthreads();
  }
  mx = s_red[0];
  __syncthreads();

  float sm = 0.f;
  for (int n = start + t; n < end; n += 256) sm += __expf(orow[n] - mx);
  s_red[t] = sm;
  __syncthreads();
  for (int o = 128; o > 0; o >>= 1) {
    if (t < o) s_red[t] += s_red[t + o];
    __syncthreads();
  }
  if (t == 0) {
    int cnt = end - start;
    float lseg = mx + __logf(s_red[0]);
    float l2 = cond_lik[b * MM + m] + lseg;  // full masked LSE of log_w
    ws_lse2[b * MM + m] = l2;
    out_cond[b * MM + m] = l2 - __logf((float)(cnt > 0 ? cnt : 1));
  }
}

// ---------------------------------------------------------------------------
// Kernel 5: post_new = log-softmax over M=8 of cond_new.  One block, width-8
// shuffle butterflies (wave32: each wave holds 4 batches x 8 models).
// ---------------------------------------------------------------------------
__global__ void k_postnew(const float* __restrict__ out_cond,
                          float* __restrict__ ws_post) {
  int t = threadIdx.x;  // 256 = 32 batches * 8 models
  float c = out_cond[t];
  float mx = c;
#pragma unroll
  for (int o = 1; o < 8; o <<= 1) mx = fmaxf(mx, __shfl_xor(mx, o, 8));
  float s = __expf(c - mx);
#pragma unroll
  for (int o = 1; o < 8; o <<= 1) s += __shfl_xor(s, o, 8);
  ws_post[t] = c - (mx + __logf(s));
}

// ---------------------------------------------------------------------------
// Kernel 6: lnw = (log_w - lse_g) + post_new[b, m_new]
// ---------------------------------------------------------------------------
__global__ void __launch_bounds__(256)
k_lnw(const float* __restrict__ out_logw, const int* __restrict__ ws_newidx,
      const float* __restrict__ ws_lse2, const float* __restrict__ ws_post,
      float* __restrict__ out_lnw) {
  const int b = (blockIdx.x << 8) / NN;  // uniform per block
  const int gid = (blockIdx.x << 8) + threadIdx.x;
  int m = ws_newidx[gid];
  out_lnw[gid] = out_logw[gid] - ws_lse2[b * MM + m] + ws_post[b * MM + m];
}

// ---------------------------------------------------------------------------
extern "C" void kernel_launch(void* const* d_in, const int* in_sizes, int n_in,
                              void* d_out, int out_size, void* d_ws,
                              size_t ws_size, hipStream_t stream) {
  (void)in_sizes; (void)n_in; (void)out_size; (void)ws_size;
  const float* x_t = (const float*)d_in[0];   // [B,N,D]
  const float* mw = (const float*)d_in[1];    // [B,N]
  const float* mp = (const float*)d_in[2];    // [B,M]
  const float* cl = (const float*)d_in[3];    // [B,M]
  const float* obs = (const float*)d_in[4];   // [B,N]
  const float* u = (const float*)d_in[5];     // [M,B,N]
  const int* midx = (const int*)d_in[6];      // [B,N]

  float* out = (float*)d_out;
  float* out_x = out;                                  // B*N*D
  float* out_logw = out_x + (size_t)BB * NN * DD;      // B*N
  float* out_lnw = out_logw + (size_t)BB * NN;         // B*N
  float* out_cond = out_lnw + (size_t)BB * NN;         // B*M

  int* ws_cum = (int*)d_ws;                            // B*M
  int* ws_spm = ws_cum + BB * MM;                      // B*M
  float* ws_lse2 = (float*)(ws_spm + BB * MM);         // B*M
  float* ws_post = ws_lse2 + BB * MM;                  // B*M
  int* ws_nidx = (int*)(ws_post + BB * MM);            // B*N
  float* ws_cdf = (float*)(ws_nidx + (size_t)BB * NN); // M*B*N  (~16.8 MB)

  k_segments<<<BB, 32, 0, stream>>>(mp, ws_cum, ws_spm);
  k_cdf<<<MM * BB, 256, 0, stream>>>(mw, midx, ws_cdf);
  k_resample<<<(BB * NN) / 256, 256, 0, stream>>>(x_t, u, cl, obs, ws_cum,
                                                  ws_cdf, ws_nidx, out_x,
                                                  out_logw);
  k_seglse<<<BB * MM, 256, 0, stream>>>(obs, cl, ws_cum, ws_spm, ws_lse2,
                                        out_cond);
  k_postnew<<<1, 256, 0, stream>>>(out_cond, ws_post);
  k_lnw<<<(BB * NN) / 256, 256, 0, stream>>>(out_logw, ws_nidx, ws_lse2,
                                             ws_post, out_lnw);
}